// PixelAugmentation_721554506283
// MI455X (gfx1250) — compile-verified
//
#include <hip/hip_runtime.h>
#include <cstdint>

// ---------------- problem constants (match reference) ----------------
constexpr int   Bc = 4, Tc = 8, Cch = 3, Hc = 600, Wc = 960;
constexpr int   Nimg = Bc * Tc;            // 32 images
constexpr int   HW   = Hc * Wc;            // 576000
constexpr int   IMG  = Cch * HW;           // 1728000 floats per image
constexpr float BLUR_PROB = 0.3f, SHARPEN_PROB = 0.3f, COLOR_PROB = 0.3f;

// geom-pass tiling
constexpr int TW = 64, TH = 20;            // output tile
constexpr int GX = Wc / TW;                // 15
constexpr int GY = Hc / TH;                // 30
constexpr int BPI = GX * GY;               // 450 blocks / image
constexpr int PR = TH + 4;                 // 24 patch rows   (y0-2 .. y0+21)
constexpr int PC = TW + 8;                 // 72 patch cols   (x0-4 .. x0+67), 16B-aligned starts
constexpr int TILEF = Cch * PR * PC;       // 5184 floats of LDS tile

static_assert(Wc % TW == 0 && Hc % TH == 0, "tiling must divide image");

// ---------------- CDNA5 data-mover paths (guarded) ----------------
#if defined(__HIP_DEVICE_COMPILE__) && __has_builtin(__builtin_amdgcn_global_load_async_to_lds_b128)
#define HAVE_ASYNC 1
#else
#define HAVE_ASYNC 0
#endif

#if defined(__HIP_DEVICE_COMPILE__) && __has_builtin(__builtin_amdgcn_tensor_load_to_lds)
#define HAVE_TDM 1
#else
#define HAVE_TDM 0
#endif

__device__ __forceinline__ float satf(float x) { return fminf(fmaxf(x, 0.0f), 1.0f); }

// jnp.pad 'reflect' (no edge repeat); halo (<=4) << dim so one reflection suffices
__device__ __forceinline__ int refl(int i, int n) {
    if (i < 0)  i = -i;
    if (i >= n) i = 2 * n - 2 - i;
    return i;
}

#if HAVE_ASYNC
typedef int v4i_ __attribute__((vector_size(16)));
typedef __attribute__((address_space(1))) v4i_* as1_v4i;   // global
typedef __attribute__((address_space(3))) v4i_* as3_v4i;   // LDS

__device__ __forceinline__ void async_g2l_b128(const float* g, const float* l) {
    // AS1 ptr: flat global VA. AS3 ptr: LDS byte offset == low 32 bits of the
    // flat shared address (ISA 10.2).
    __builtin_amdgcn_global_load_async_to_lds_b128(
        (as1_v4i)(uintptr_t)g,
        (as3_v4i)(uint32_t)(uintptr_t)l,
        0, 0);
}
__device__ __forceinline__ void wait_async0() {
#if __has_builtin(__builtin_amdgcn_s_wait_asynccnt)
    __builtin_amdgcn_s_wait_asynccnt(0);
#else
    asm volatile("s_wait_asynccnt 0" ::: "memory");
#endif
}
#endif

#if HAVE_TDM
typedef unsigned int tdm_v4u __attribute__((ext_vector_type(4)));
typedef int          tdm_v8i __attribute__((ext_vector_type(8)));
typedef int          tdm_v4i __attribute__((ext_vector_type(4)));

__device__ __forceinline__ void wait_tensor0() {
#if __has_builtin(__builtin_amdgcn_s_wait_tensorcnt)
    __builtin_amdgcn_s_wait_tensorcnt(0);
#else
    asm volatile("s_wait_tensorcnt 0" ::: "memory");
#endif
}

// One TENSOR_LOAD_TO_LDS: 3D tile (x=PC cols, y=PR rows, z=Cch channels) from
// global (row stride Wc, channel stride HW) -> contiguous LDS [ch][row][PC].
// D# layout per CDNA5 ISA §8.3-8.6.
__device__ __forceinline__ void tdm_load_tile(const float* gsrc, const float* lds)
{
    const uint64_t ga = (uint64_t)(uintptr_t)gsrc;   // 57-bit global byte address
    const uint32_t la = (uint32_t)(uintptr_t)lds;    // LDS byte offset (flat low 32)

    tdm_v4u g0;
    g0.x = 1u;                                       // count=1, user mode, no gather
    g0.y = la;                                       // lds_addr [63:32]
    g0.z = (uint32_t)ga;                             // global_addr[31:0]
    g0.w = (uint32_t)((ga >> 32) & 0x01FFFFFFu)      // global_addr[56:32]
         | (2u << 30);                               // type = 2 ("image")

    const uint32_t t0 = (uint32_t)PC;                // tensor_dim0 == tile_dim0 (no OOB)
    const uint32_t t1 = (uint32_t)PR;                // tensor_dim1 == tile_dim1
    const uint64_t s1 = (uint64_t)HW;                // dim1 (channel) stride, elements

    tdm_v8i g1;
    g1.s0 = (int)(2u << 16);                         // wg_mask=0 | data_size=2 (4B)
    g1.s1 = (int)((t0 & 0xFFFFu) << 16);             // atomic_bar=0 | tensor_dim0.lo16
    g1.s2 = (int)((t0 >> 16) | ((t1 & 0xFFFFu) << 16));  // dim0.hi16 | dim1.lo16
    g1.s3 = (int)((t1 >> 16) | ((uint32_t)PC << 16));    // dim1.hi16 | tile_dim0=72
    g1.s4 = (int)((uint32_t)PR | ((uint32_t)Cch << 16)); // tile_dim1=24 | tile_dim2=3
    g1.s5 = (int)(uint32_t)Wc;                       // tensor_dim0_stride[31:0] = 960
    g1.s6 = (int)(((uint32_t)(s1 & 0xFFFFu)) << 16); // dim0_stride[47:32]=0 | dim1_stride.lo16
    g1.s7 = (int)(uint32_t)(s1 >> 16);               // tensor_dim1_stride[47:16]

    tdm_v4i g2;
    g2.x = (int)(uint32_t)Cch;                       // tensor_dim2 = 3
    g2.y = 0;                                        // tensor_dim3 (unused)
    g2.z = 0;                                        // tensor_dim2_stride.lo (unused: tile_dim3=0)
    g2.w = 0;                                        // dim2_stride.hi | tile_dim3=0

    tdm_v4i g3; g3.x = 0; g3.y = 0; g3.z = 0; g3.w = 0;

#if __clang_major__ >= 23
    // clang-23 form: extra int32x8 group (unused VADDR4 per ISA: "set to NULL")
    tdm_v8i g4;
    g4.s0 = 0; g4.s1 = 0; g4.s2 = 0; g4.s3 = 0;
    g4.s4 = 0; g4.s5 = 0; g4.s6 = 0; g4.s7 = 0;
    __builtin_amdgcn_tensor_load_to_lds(g0, g1, g2, g3, g4, 0);
#else
    __builtin_amdgcn_tensor_load_to_lds(g0, g1, g2, g3, 0);
#endif
}
#endif

// =====================================================================
// Pass 1: blur / sharpen / identity (per-image uniform), write d_out,
//         accumulate gray(clip(out*bf)) per block into ws (deterministic).
// =====================================================================
__global__ __launch_bounds__(256)
void geom_kernel(const float* __restrict__ img, const float* __restrict__ rv_,
                 const float* __restrict__ stdu_, const float* __restrict__ shu_,
                 const float* __restrict__ jit_, float* __restrict__ out,
                 float* __restrict__ pw)
{
    __shared__ float smem[TILEF + 256];          // tile + reduction scratch
    const int tid = threadIdx.x;
    const int n  = blockIdx.z;
    const int x0 = blockIdx.x * TW;
    const int y0 = blockIdx.y * TH;
    const float* src = img + (size_t)n * IMG;
    float*       dst = out + (size_t)n * IMG;

    const float rv  = rv_[n];
    const float bf  = 1.0f + 0.3f * (2.0f * jit_[n * 4 + 0] - 1.0f);
    const float shf = 1.0f + 4.0f * shu_[n];     // SH_LO + (SH_HI-SH_LO)*u

    // Gaussian 1D weights for std = 0.1 + 1.6*u, taps at -2..2
    float w0, w1, w2;
    {
        const float inv = 1.0f / (0.1f + 1.6f * stdu_[n]);
        const float e1 = expf(-0.5f * inv * inv);
        const float e2 = expf(-2.0f * inv * inv);
        const float r  = 1.0f / (1.0f + 2.0f * e1 + 2.0f * e2);
        w0 = e2 * r; w1 = e1 * r; w2 = r;        // symmetric: w3=w1, w4=w0
    }
    const float w[5] = {w0, w1, w2, w1, w0};

    // ---- stage (3 x 24 x 72) input patch into LDS ----
    const bool interior = (x0 >= 4) && (x0 + PC - 4 <= Wc) && (y0 >= 2) && (y0 + PR - 2 <= Hc);
#if HAVE_TDM
    if (interior) {                              // single TDM DMA for the whole tile
        if (tid < 32) {                          // one wave issues; TDM ignores EXEC
            tdm_load_tile(src + ((size_t)(y0 - 2)) * Wc + (x0 - 4), smem);
            wait_tensor0();                      // TENSORcnt -> 0 in issuing wave
        }
    } else
#elif HAVE_ASYNC
    if (interior) {                              // 1296 b128 async DMA chunks
        for (int c = tid; c < Cch * PR * (PC / 4); c += 256) {
            const int ch  = c / (PR * (PC / 4));
            const int rem = c - ch * (PR * (PC / 4));
            const int r   = rem / (PC / 4);
            const int cc  = rem - r * (PC / 4);
            const float* g = src + ((size_t)ch * Hc + (y0 - 2 + r)) * Wc + (x0 - 4) + cc * 4;
            async_g2l_b128(g, &smem[(ch * PR + r) * PC + cc * 4]);
        }
        wait_async0();
    } else
#endif
    {
        (void)interior;                          // generic fill with reflect padding
        for (int idx = tid; idx < TILEF; idx += 256) {
            const int ch  = idx / (PR * PC);
            const int rem = idx - ch * (PR * PC);
            const int r   = rem / PC;
            const int c   = rem - r * PC;
            const int gy = refl(y0 - 2 + r, Hc);
            const int gx = refl(x0 - 4 + c, Wc);
            smem[idx] = src[((size_t)ch * Hc + gy) * Wc + gx];
        }
    }
    __syncthreads();

    const bool doBlur  = (rv < BLUR_PROB);
    const bool doSharp = (!doBlur) && (rv < BLUR_PROB + SHARPEN_PROB);

    float lsum = 0.0f;
#pragma unroll
    for (int k = 0; k < (TW * TH) / 256; ++k) {  // 5 pixels / thread
        const int p  = tid + k * 256;
        const int ty = p >> 6, tx = p & 63;
        const int gy = y0 + ty, gx = x0 + tx;
        float o[3];
        if (doBlur) {
#pragma unroll
            for (int ch = 0; ch < 3; ++ch) {
                float acc = 0.0f;
#pragma unroll
                for (int i = 0; i < 5; ++i) {
                    const float* row = &smem[(ch * PR + ty + i) * PC + tx + 2];
                    const float rs = w[0]*row[0] + w[1]*row[1] + w[2]*row[2] + w[3]*row[3] + w[4]*row[4];
                    acc = fmaf(w[i], rs, acc);
                }
                o[ch] = acc;
            }
        } else if (doSharp) {
            const bool inter = (gy >= 1) && (gy <= Hc - 2) && (gx >= 1) && (gx <= Wc - 2);
#pragma unroll
            for (int ch = 0; ch < 3; ++ch) {
                const float c = smem[(ch * PR + ty + 2) * PC + tx + 4];
                float deg = c;
                if (inter) {
                    float s9 = 0.0f;
#pragma unroll
                    for (int i = 0; i < 3; ++i) {
                        const float* row = &smem[(ch * PR + ty + 1 + i) * PC + tx + 3];
                        s9 += row[0] + row[1] + row[2];
                    }
                    s9 += 4.0f * c;              // center weight 5/13 total
                    deg = satf(s9 * (1.0f / 13.0f));
                }
                o[ch] = satf(shf * c + (1.0f - shf) * deg);
            }
        } else {
#pragma unroll
            for (int ch = 0; ch < 3; ++ch)
                o[ch] = smem[(ch * PR + ty + 2) * PC + tx + 4];
        }

        // gray of brightness-clipped geom result (feeds contrast mean)
        const float jr = satf(o[0] * bf), jg = satf(o[1] * bf), jb = satf(o[2] * bf);
        lsum += 0.299f * jr + 0.587f * jg + 0.114f * jb;

        const size_t po = (size_t)gy * Wc + gx;
        dst[0 * HW + po] = o[0];
        dst[1 * HW + po] = o[1];
        dst[2 * HW + po] = o[2];
    }

    // deterministic block reduction -> unique ws slot (no float atomics)
    float* red = smem + TILEF;
    red[tid] = lsum;
    __syncthreads();
    for (int s = 128; s > 0; s >>= 1) {
        if (tid < s) red[tid] += red[tid + s];
        __syncthreads();
    }
    if (tid == 0) pw[n * BPI + blockIdx.y * GX + blockIdx.x] = red[0];
}

// =====================================================================
// Pass 1b: reduce 450 partials/image -> mean (fixed order, deterministic)
// =====================================================================
__global__ __launch_bounds__(256)
void mean_kernel(const float* __restrict__ pw, float* __restrict__ means)
{
    __shared__ float red[256];
    const int n = blockIdx.x, tid = threadIdx.x;
    float s = 0.0f;
    if (tid < BPI)       s  = pw[n * BPI + tid];
    if (tid + 256 < BPI) s += pw[n * BPI + tid + 256];
    red[tid] = s;
    __syncthreads();
    for (int st = 128; st > 0; st >>= 1) {
        if (tid < st) red[tid] += red[tid + st];
        __syncthreads();
    }
    if (tid == 0) means[n] = red[0] * (1.0f / (float)HW);
}

// =====================================================================
// Pass 2: color jitter in place on d_out, block-uniform skip when !cmask
// =====================================================================
__global__ __launch_bounds__(256)
void color_kernel(const float* __restrict__ jit_, const float* __restrict__ cu_,
                  const float* __restrict__ means, float* __restrict__ out)
{
    const int n = blockIdx.z;
    if (cu_[n] >= COLOR_PROB) return;            // ~70% of images: zero traffic
    const int p4 = blockIdx.x * 256 + threadIdx.x;
    if (p4 >= HW / 4) return;

    float* base = out + (size_t)n * IMG + (size_t)p4 * 4;
    const float bf = 1.0f + 0.3f * (2.0f * jit_[n * 4 + 0] - 1.0f);
    const float cf = 1.0f + 0.3f * (2.0f * jit_[n * 4 + 1] - 1.0f);
    const float sf = 1.0f + 0.3f * (2.0f * jit_[n * 4 + 2] - 1.0f);
    const float hf = 0.1f * (2.0f * jit_[n * 4 + 3] - 1.0f);
    const float mean = means[n];

    __builtin_prefetch(base + HW, 0, 0);         // global_prefetch_b8
    __builtin_prefetch(base + 2 * HW, 0, 0);

    float4 R  = *(const float4*)(base);
    float4 G  = *(const float4*)(base + HW);
    float4 Bv = *(const float4*)(base + 2 * HW);
    float* rp = (float*)&R; float* gp = (float*)&G; float* bp = (float*)&Bv;

#pragma unroll
    for (int k = 0; k < 4; ++k) {
        float r = satf(rp[k] * bf), g = satf(gp[k] * bf), b = satf(bp[k] * bf);
        r = satf(cf * r + (1.0f - cf) * mean);
        g = satf(cf * g + (1.0f - cf) * mean);
        b = satf(cf * b + (1.0f - cf) * mean);
        const float gray = 0.299f * r + 0.587f * g + 0.114f * b;
        r = satf(sf * r + (1.0f - sf) * gray);
        g = satf(sf * g + (1.0f - sf) * gray);
        b = satf(sf * b + (1.0f - sf) * gray);

        // rgb2hsv (mirrors reference's where-chain exactly)
        const float maxc = fmaxf(r, fmaxf(g, b));
        const float minc = fminf(r, fminf(g, b));
        const float cr   = maxc - minc;
        const bool  eq   = (maxc == minc);
        const float s    = cr / (eq ? 1.0f : maxc);
        const float crd  = eq ? 1.0f : cr;
        const float rc = (maxc - r) / crd, gc = (maxc - g) / crd, bc = (maxc - b) / crd;
        float h = (maxc == r) ? (bc - gc)
                : (maxc == g) ? (2.0f + rc - bc)
                              : (4.0f + gc - rc);
        h *= (1.0f / 6.0f);
        h -= floorf(h);                          // python % 1.0
        h += hf;
        h -= floorf(h);

        // hsv2rgb
        const float v  = maxc;
        const float hh = h * 6.0f;
        const float fi = floorf(hh);
        const float f  = hh - fi;
        const int   i  = ((int)fi) % 6;
        const float p = satf(v * (1.0f - s));
        const float q = satf(v * (1.0f - s * f));
        const float t = satf(v * (1.0f - s * (1.0f - f)));
        float rr, gg, bb;
        switch (i) {
            case 0: rr = v; gg = t; bb = p; break;
            case 1: rr = q; gg = v; bb = p; break;
            case 2: rr = p; gg = v; bb = t; break;
            case 3: rr = p; gg = q; bb = v; break;
            case 4: rr = t; gg = p; bb = v; break;
            default: rr = v; gg = p; bb = q; break;
        }
        rp[k] = rr; gp[k] = gg; bp[k] = bb;
    }
    *(float4*)(base)          = R;
    *(float4*)(base + HW)     = G;
    *(float4*)(base + 2 * HW) = Bv;
}

// =====================================================================
extern "C" void kernel_launch(void* const* d_in, const int* in_sizes, int n_in,
                              void* d_out, int out_size, void* d_ws, size_t ws_size,
                              hipStream_t stream)
{
    (void)in_sizes; (void)n_in; (void)out_size; (void)ws_size;
    const float* image      = (const float*)d_in[0];
    const float* rand_value = (const float*)d_in[1];
    const float* std_u      = (const float*)d_in[2];
    const float* sharpen_u  = (const float*)d_in[3];
    const float* color_u    = (const float*)d_in[4];
    const float* jitter_u   = (const float*)d_in[5];
    float* out = (float*)d_out;

    float* pw    = (float*)d_ws;                 // Nimg*BPI partial sums
    float* means = pw + Nimg * BPI;              // Nimg means

    dim3 g1(GX, GY, Nimg);
    geom_kernel<<<g1, 256, 0, stream>>>(image, rand_value, std_u, sharpen_u,
                                        jitter_u, out, pw);
    mean_kernel<<<Nimg, 256, 0, stream>>>(pw, means);
    dim3 g2((HW / 4 + 255) / 256, 1, Nimg);
    color_kernel<<<g2, 256, 0, stream>>>(jitter_u, color_u, means, out);
}